// Sinkhorn_38981123178755
// MI455X (gfx1250) — compile-verified
//
#include <hip/hip_runtime.h>
#include <cstdint>

// Sinkhorn (log-domain reference) reformulated as dual-potential updates:
//   K = exp(s/temp);  v[m] = 1/sum_n K*u;  u[n] = 1/sum_m K*v;  out = K*u*v
// Sequence matching the reference (init col; i=0..14 alternating row/col):
//   col(u=1); 7 x (row, col); final row (fused with out = K*v*u write).
// Memory-bound: 16 scans x 256MB + 256MB write ~ 4.3GB ~ 185us @ 23.3TB/s.
// exp computed as exp2(x * log2e/temp) -> single v_exp_f32 + one v_mul.

#if __has_builtin(__builtin_amdgcn_global_load_async_to_lds_b128) && \
    __has_builtin(__builtin_amdgcn_s_wait_asynccnt)
#define USE_ASYNC_LDS 1
#endif

#define LOG2E 1.4426950408889634f

typedef int v4i __attribute__((ext_vector_type(4)));
typedef __attribute__((address_space(1))) v4i* gv4i_ptr;  // global (device) AS
typedef __attribute__((address_space(3))) v4i* lv4i_ptr;  // LDS AS

__device__ __forceinline__ gv4i_ptr as_global_v4i(const void* p) {
  // int->ptr cast avoids addrspacecast restrictions; drops const for builtin
  return (gv4i_ptr)(uint64_t)p;
}
__device__ __forceinline__ lv4i_ptr as_lds_v4i(void* p) {
  // generic LDS address: low 32 bits are the LDS byte offset (aperture scheme)
  return (lv4i_ptr)(uint32_t)(uint64_t)p;
}

// Copy one float4 from global to LDS: async DMA path on CDNA5, plain fallback.
__device__ __forceinline__ void stage_f4(const float4* __restrict__ gsrc,
                                         float4* ldst) {
#ifdef USE_ASYNC_LDS
  __builtin_amdgcn_global_load_async_to_lds_b128(as_global_v4i(gsrc),
                                                 as_lds_v4i(ldst), 0, 0);
#else
  *ldst = *gsrc;
#endif
}
__device__ __forceinline__ void stage_wait() {
#ifdef USE_ASYNC_LDS
  __builtin_amdgcn_s_wait_asynccnt(0);
#endif
}

// hardware exp2 (v_exp_f32); inputs here are O(10), well within range
__device__ __forceinline__ float hw_exp2(float x) {
  return __builtin_amdgcn_exp2f(x);
}

// ---------------- Row pass: u[n] = 1 / sum_m exp(s/temp) * v[m] --------------
// grid = B*1024/8 blocks, 256 threads (8 wave32's), one wave per row.
// FINAL: also writes out[n,m] = exp(s/temp)*v[m]*u[n] from registers (no
// re-read of s).
template <bool FINAL>
__global__ void __launch_bounds__(256) sinkhorn_row(
    const float* __restrict__ s, const float* __restrict__ temp,
    const float* __restrict__ v, float* __restrict__ u,
    float* __restrict__ out) {
  __shared__ float4 vs[256];  // v[b, 0..1023]

  const int block_row0 = blockIdx.x * 8;
  const int b = block_row0 >> 10;  // 1024 rows per batch, 128 blocks per batch

  stage_f4(((const float4*)(v + ((size_t)b << 10))) + threadIdx.x,
           &vs[threadIdx.x]);
  stage_wait();
  __syncthreads();

  const float c = LOG2E / temp[0];  // exp(x/temp) == exp2(x*c)
  const int wave = threadIdx.x >> 5;
  const int lane = threadIdx.x & 31;
  const long long row = (long long)block_row0 + wave;
  const float4* __restrict__ srow = (const float4*)(s + (row << 10));

  float acc = 0.0f;
  float4 pv[8];  // K*v values kept in registers for the fused final write
#pragma unroll
  for (int k = 0; k < 8; ++k) {
    const int idx = lane + (k << 5);  // float4 index; covers m = 4*idx..4*idx+3
    const float4 x = srow[idx];
    const float4 vv = vs[idx];
    const float p0 = hw_exp2(x.x * c) * vv.x;
    const float p1 = hw_exp2(x.y * c) * vv.y;
    const float p2 = hw_exp2(x.z * c) * vv.z;
    const float p3 = hw_exp2(x.w * c) * vv.w;
    acc += (p0 + p1) + (p2 + p3);
    if (FINAL) {
      pv[k].x = p0; pv[k].y = p1; pv[k].z = p2; pv[k].w = p3;
    }
  }
  // wave32 butterfly reduction — all lanes end with the row sum
#pragma unroll
  for (int off = 16; off > 0; off >>= 1) acc += __shfl_xor(acc, off, 32);

  const float uu = 1.0f / acc;
  if (FINAL) {
    float4* __restrict__ orow = (float4*)(out + (row << 10));
#pragma unroll
    for (int k = 0; k < 8; ++k) {
      const int idx = lane + (k << 5);
      float4 o;
      o.x = pv[k].x * uu; o.y = pv[k].y * uu;
      o.z = pv[k].z * uu; o.w = pv[k].w * uu;
      orow[idx] = o;
    }
  } else if (lane == 0) {
    u[row] = uu;
  }
}

// ---------------- Col pass: v[m] = 1 / sum_n exp(s/temp) * u[n] --------------
// grid = B*4 blocks, 1024 threads: col = tid&255 (256-col tile), chunk = tid>>8
// (4-way split of the n range for occupancy); LDS combine at the end.
__global__ void __launch_bounds__(1024) sinkhorn_col(
    const float* __restrict__ s, const float* __restrict__ temp,
    const float* __restrict__ u, float* __restrict__ v, int use_u) {
  __shared__ float us[1024];
  __shared__ float part[4][256];

  const int b = blockIdx.x >> 2;
  const int mt = (blockIdx.x & 3) << 8;

  if (use_u) {
    if (threadIdx.x < 256)
      stage_f4(((const float4*)(u + ((size_t)b << 10))) + threadIdx.x,
               ((float4*)us) + threadIdx.x);
    stage_wait();
  } else {
    us[threadIdx.x] = 1.0f;  // initial normalization: u == 1
  }
  __syncthreads();

  const float c = LOG2E / temp[0];  // exp(x/temp) == exp2(x*c)
  const int col = threadIdx.x & 255;
  const int chunk = threadIdx.x >> 8;
  const int m = mt + col;
  const int n0 = chunk << 8;
  const float* __restrict__ sp =
      s + ((long long)b * 1024 + n0) * 1024 + m;

  float a0 = 0.f, a1 = 0.f, a2 = 0.f, a3 = 0.f;
  for (int n = 0; n < 256; n += 4) {
    // strided (4KB) column walk: software-prefetch ahead (speculative, safe)
    __builtin_prefetch(sp + ((size_t)(n + 24) << 10), 0, 1);
    const float x0 = sp[(size_t)(n + 0) << 10];
    const float x1 = sp[(size_t)(n + 1) << 10];
    const float x2 = sp[(size_t)(n + 2) << 10];
    const float x3 = sp[(size_t)(n + 3) << 10];
    a0 += hw_exp2(x0 * c) * us[n0 + n + 0];
    a1 += hw_exp2(x1 * c) * us[n0 + n + 1];
    a2 += hw_exp2(x2 * c) * us[n0 + n + 2];
    a3 += hw_exp2(x3 * c) * us[n0 + n + 3];
  }
  part[chunk][col] = (a0 + a1) + (a2 + a3);
  __syncthreads();

  if (threadIdx.x < 256) {
    const float tot =
        (part[0][col] + part[1][col]) + (part[2][col] + part[3][col]);
    v[((size_t)b << 10) + m] = 1.0f / tot;
  }
}

extern "C" void kernel_launch(void* const* d_in, const int* in_sizes, int n_in,
                              void* d_out, int out_size, void* d_ws,
                              size_t ws_size, hipStream_t stream) {
  const float* s = (const float*)d_in[0];
  const float* temp = (const float*)d_in[1];
  const int B = in_sizes[0] / (1024 * 1024);  // 64

  // workspace: u[B*1024], v[B*1024] (fully rewritten every call)
  float* u = (float*)d_ws;
  float* v = u + (size_t)B * 1024;
  float* out = (float*)d_out;

  const dim3 cgrid(B * 4), cblk(1024);
  const dim3 rgrid(B * 128), rblk(256);

  // init: axis=1 (columns), u == 1
  sinkhorn_col<<<cgrid, cblk, 0, stream>>>(s, temp, u, v, 0);
  // i = 0..13: alternating axis=2 (rows) / axis=1 (columns)
  for (int i = 0; i < 7; ++i) {
    sinkhorn_row<false><<<rgrid, rblk, 0, stream>>>(s, temp, v, u, nullptr);
    sinkhorn_col<<<cgrid, cblk, 0, stream>>>(s, temp, u, v, 1);
  }
  // i = 14: final row pass, fused with out = K*v*u
  sinkhorn_row<true><<<rgrid, rblk, 0, stream>>>(s, temp, v, u, out);
}